// DSNAS_v_11579231830853
// MI455X (gfx1250) — compile-verified
//
#include <hip/hip_runtime.h>
#include <math.h>

// ---------------------------------------------------------------------------
// DSNAS forward, algebraically folded:
//   out[b,k] = sum_{i,d} e*Wlin + sum_{i<j,d} |e_i-e_j|*Sd + e_i*e_j*Q
//            + sum_{i,d} e^2*Qd
// Linear term runs on the matrix core (v_wmma_f32_16x16x4_f32, exact f32).
// ---------------------------------------------------------------------------

#define NC 22      // columns
#define EN 12      // embeddings per column
#define DD 64      // feature dim
#define BB 2048    // batch
#define NP 231     // NC*(NC-1)/2 unordered pairs
#define ROW (NC*DD)  // 1408 features per batch row

typedef float v2f __attribute__((ext_vector_type(2)));
typedef float v8f __attribute__((ext_vector_type(8)));

// W_ops[i][j][op][d][k] : op 0=plus 1=mul 2=max 3=min
__device__ __forceinline__ float WOP(const float* W, int i, int j, int op, int d, int k) {
    return W[((((i * NC + j) * 4 + op) * DD + d) * 2 + k)];
}
// W_cat[i][j][c][k], c in [0,128)
__device__ __forceinline__ float WCAT(const float* W, int i, int j, int c, int k) {
    return W[(((i * NC + j) * 128 + c) * 2 + k)];
}

// ---------------------------------------------------------------------------
// Kernel 1: fold mixture weights + symmetry into small tables.
//   blocks 0..230  -> pair p=(i<j): Sd[p][d][k], Q[p][d][k]
//   blocks 231..252-> column i:     Qd[i][d][k], WLpad[(i*64+d)*16 + n]
// blockDim = 64 (d = threadIdx.x)
// ---------------------------------------------------------------------------
__global__ void k_prep(const float* __restrict__ Wops, const float* __restrict__ Wcat,
                       const float* __restrict__ la,
                       float* __restrict__ Sd, float* __restrict__ Q,
                       float* __restrict__ Qd, float* __restrict__ WL) {
    const int d = threadIdx.x;
    // softmax over 5 logits (redundant per thread; trivial)
    float a0 = la[0], a1 = la[1], a2 = la[2], a3 = la[3], a4 = la[4];
    float mx = fmaxf(fmaxf(fmaxf(a0, a1), fmaxf(a2, a3)), a4);
    float e0 = expf(a0 - mx), e1 = expf(a1 - mx), e2 = expf(a2 - mx),
          e3 = expf(a3 - mx), e4 = expf(a4 - mx);
    float inv = 1.0f / (e0 + e1 + e2 + e3 + e4);
    const float w_plus = e0 * inv, w_mul = e1 * inv, w_max = e2 * inv,
                w_min = e3 * inv, w_cat = e4 * inv;

    int p = blockIdx.x;
    if (p < NP) {
        // recover (i, j) with i<j from linear pair index
        int i = 0, rem = p;
        while (rem >= NC - 1 - i) { rem -= NC - 1 - i; ++i; }
        int j = i + 1 + rem;
        for (int k = 0; k < 2; ++k) {
            float smax = WOP(Wops, i, j, 2, d, k) + WOP(Wops, j, i, 2, d, k);
            float smin = WOP(Wops, i, j, 3, d, k) + WOP(Wops, j, i, 3, d, k);
            float smul = WOP(Wops, i, j, 1, d, k) + WOP(Wops, j, i, 1, d, k);
            Sd[p * 128 + d * 2 + k] = 0.5f * (w_max * smax - w_min * smin);
            Q [p * 128 + d * 2 + k] = w_mul * smul;
        }
    } else {
        int i = p - NP;  // 0..21
        float lin[2];
        for (int k = 0; k < 2; ++k) {
            float acc = 0.0f;
            for (int o = 0; o < NC; ++o) {
                acc += w_plus * (WOP(Wops, i, o, 0, d, k) + WOP(Wops, o, i, 0, d, k));
                acc += 0.5f * w_max * (WOP(Wops, i, o, 2, d, k) + WOP(Wops, o, i, 2, d, k));
                acc += 0.5f * w_min * (WOP(Wops, i, o, 3, d, k) + WOP(Wops, o, i, 3, d, k));
                acc += w_cat * (WCAT(Wcat, i, o, d, k) + WCAT(Wcat, o, i, DD + d, k));
            }
            lin[k] = acc;
            Qd[i * 128 + d * 2 + k] = w_mul * WOP(Wops, i, i, 1, d, k);
        }
        // WMMA B-matrix row (K = i*64+d), N padded 2 -> 16 with zeros
        float* wrow = WL + (i * DD + d) * 16;
        wrow[0] = lin[0];
        wrow[1] = lin[1];
        for (int n = 2; n < 16; ++n) wrow[n] = 0.0f;
    }
}

// ---------------------------------------------------------------------------
// Kernel 2: embeddings.  E[b][i*64+d] = mu + softplus(sigma)*v[b,d]*0.01
// grid (B, NC), block 64
// ---------------------------------------------------------------------------
__global__ void k_embed(const int* __restrict__ idx, const float* __restrict__ em,
                        const float* __restrict__ es, const float* __restrict__ v,
                        float* __restrict__ E) {
    const int b = blockIdx.x, i = blockIdx.y, d = threadIdx.x;
    const int id = idx[i * BB + b];
    const float m = em[(i * EN + id) * DD + d];
    const float s = es[(i * EN + id) * DD + d];
    const float sp = (s > 20.0f) ? s : log1pf(expf(s));
    E[b * ROW + i * DD + d] = fmaf(sp * v[b * DD + d], 0.01f, m);
}

// ---------------------------------------------------------------------------
// Kernel 3: linear term via matrix core.  Lin[B x 2] = E[B x 1408] @ WL[1408 x 16(:2)]
// One wave per 16-row M-tile; K loop in steps of 4 using V_WMMA_F32_16X16X4_F32.
// A layout (ISA 7.12.2, 32-bit 16x4): lanes 0-15 hold K={0,1}, lanes 16-31 K={2,3},
// M = lane&15, one K per VGPR. B (4x16) assumed to mirror the same K-half split:
// VGPR t -> row K = t + 2*(lane>>4), N = lane&15. D: VGPR r -> M = r + 8*(lane>>4),
// N = lane&15.
// block 256 (8 waves), grid 16  -> 128 tiles = 2048 rows
// ---------------------------------------------------------------------------
__global__ void k_lin_wmma(const float* __restrict__ E, const float* __restrict__ WL,
                           float* __restrict__ Lin) {
    const int lane = threadIdx.x & 31;
    const int wave = blockIdx.x * (blockDim.x >> 5) + (threadIdx.x >> 5);
    const int b0 = wave * 16;
    const int mrow = b0 + (lane & 15);
    const int khalf = (lane >> 4) << 1;  // 0 for lanes 0-15, 2 for lanes 16-31
    const int n = lane & 15;

    v8f c = {};
    const float* erow = E + mrow * ROW + khalf;
    for (int kk = 0; kk < ROW; kk += 4) {
        v2f a = *(const v2f*)(erow + kk);
        v2f bf;
        bf[0] = WL[(kk + khalf + 0) * 16 + n];
        bf[1] = WL[(kk + khalf + 1) * 16 + n];
        c = __builtin_amdgcn_wmma_f32_16x16x4_f32(false, a, false, bf,
                                                  (short)0, c, false, false);
    }
    // only N=0,1 carry the real output
    if (n < 2) {
        const int mbase = b0 + ((lane >> 4) << 3);
#pragma unroll
        for (int r = 0; r < 8; ++r) Lin[(mbase + r) * 2 + n] = c[r];
    }
}

// ---------------------------------------------------------------------------
// Kernel 4: pairwise terms.  One wave per batch row; lane l owns d = {2l, 2l+1};
// e stays in 44 VGPRs; weight streams are coalesced float4 (16B/lane = 512B/wave,
// L2-resident ~270KB). Fully unrolled 231-pair loop, then wave32 shfl reduction.
// block 256 (8 waves), grid 64 -> 512 waves, 4 rows each.
// ---------------------------------------------------------------------------
__global__ void k_pair(const float* __restrict__ E, const float* __restrict__ Sd,
                       const float* __restrict__ Q, const float* __restrict__ Qd,
                       const float* __restrict__ Lin, float* __restrict__ out) {
    const int lane = threadIdx.x & 31;
    const int gw = blockIdx.x * 8 + (threadIdx.x >> 5);
    const int nw = gridDim.x * 8;

    for (int b = gw; b < BB; b += nw) {
        float2 e[NC];
        const float* Eb = E + b * ROW + 2 * lane;
#pragma unroll
        for (int i = 0; i < NC; ++i) e[i] = *(const float2*)(Eb + i * DD);

        float acc0 = 0.0f, acc1 = 0.0f;

        // diagonal quadratic (mul term, i == j)
#pragma unroll
        for (int i = 0; i < NC; ++i) {
            float4 q = *(const float4*)(Qd + i * 128 + lane * 4);
            float xx = e[i].x * e[i].x, yy = e[i].y * e[i].y;
            acc0 += xx * q.x + yy * q.z;
            acc1 += xx * q.y + yy * q.w;
        }

        // unordered pairs: |diff| (max/min antisym) and product (mul off-diag)
        int p = 0;
#pragma unroll
        for (int i = 0; i < NC; ++i) {
#pragma unroll
            for (int j = i + 1; j < NC; ++j) {
                float4 sd = *(const float4*)(Sd + p * 128 + lane * 4);
                float4 qq = *(const float4*)(Q + p * 128 + lane * 4);
                float dx = fabsf(e[i].x - e[j].x);
                float dy = fabsf(e[i].y - e[j].y);
                float px = e[i].x * e[j].x;
                float py = e[i].y * e[j].y;
                acc0 += dx * sd.x + dy * sd.z + px * qq.x + py * qq.z;
                acc1 += dx * sd.y + dy * sd.w + px * qq.y + py * qq.w;
                ++p;
            }
        }

        // wave32 butterfly reduction
#pragma unroll
        for (int off = 16; off; off >>= 1) {
            acc0 += __shfl_xor(acc0, off, 32);
            acc1 += __shfl_xor(acc1, off, 32);
        }
        if (lane == 0) {
            out[b * 2 + 0] = acc0 + Lin[b * 2 + 0];
            out[b * 2 + 1] = acc1 + Lin[b * 2 + 1];
        }
    }
}

// ---------------------------------------------------------------------------
// Host launcher. ws layout (floats):
//   Sd 29568 | Q 29568 | Qd 2816 | WL 22528 | Lin 4096 | E 2883584  (~11.9 MB)
// All buffers fully written before read; no cross-call state.
// ---------------------------------------------------------------------------
extern "C" void kernel_launch(void* const* d_in, const int* in_sizes, int n_in,
                              void* d_out, int out_size, void* d_ws, size_t ws_size,
                              hipStream_t stream) {
    const int*   idx  = (const int*)  d_in[0];
    const float* em   = (const float*)d_in[1];
    const float* es   = (const float*)d_in[2];
    const float* v    = (const float*)d_in[3];
    const float* Wops = (const float*)d_in[4];
    const float* Wcat = (const float*)d_in[5];
    const float* la   = (const float*)d_in[6];
    float* out = (float*)d_out;

    float* ws  = (float*)d_ws;
    float* Sd  = ws;
    float* Q   = Sd + NP * 128;        // 29568
    float* Qd  = Q + NP * 128;         // 29568
    float* WL  = Qd + NC * 128;        // 2816
    float* Lin = WL + ROW * 16;        // 22528
    float* E   = Lin + BB * 2;         // 4096

    k_prep<<<NP + NC, 64, 0, stream>>>(Wops, Wcat, la, Sd, Q, Qd, WL);
    k_embed<<<dim3(BB, NC), 64, 0, stream>>>(idx, em, es, v, E);
    k_lin_wmma<<<16, 256, 0, stream>>>(E, WL, Lin);
    k_pair<<<64, 256, 0, stream>>>(E, Sd, Q, Qd, Lin, out);
}